// RGCNConv_39487929319722
// MI455X (gfx1250) — compile-verified
//
#include <hip/hip_runtime.h>
#include <hip/hip_bf16.h>

typedef __attribute__((ext_vector_type(2))) float v2f;
typedef __attribute__((ext_vector_type(8))) float v8f;

#define NN   100000   // nodes
#define SS   50       // relations
#define BB   30       // bases
#define OUTD 16       // output dim
#define EE   20000    // edges per relation
#define NPR  (NN/SS)  // 2000: node-block size per relation (torch-quirk indexing)
#define BPAD 32       // K padded 30 -> 32 (8 wmma k-steps of 4)
#define SPAD 64       // M padded 50 -> 64 (4 wmma m-tiles of 16)
#define CRCAP 25      // max relations per chunk: 25*6.4MB = 160MB Vblk <= 192MB L2

__global__ __launch_bounds__(256) void rgcn_zero(float* __restrict__ p, int n) {
  int i = blockIdx.x * 256 + threadIdx.x;
  if (i < n) p[i] = 0.0f;
}

__global__ __launch_bounds__(256) void rgcn_relu(float* __restrict__ p, int n) {
  int i = blockIdx.x * 256 + threadIdx.x;
  if (i < n) p[i] = fmaxf(p[i], 0.0f);
}

// Phase 1: per node n (chunk-local cn), compute V_slice[cn][s][o] =
//   sum_b Wc[s][b] * W[(b*NN + n)*16 + o]   via  (64x32)@(32x16) WMMA f32.
// One wave per node. A (zero-padded W_comp) is hoisted to registers, so the
// k-step-7 B rows for b=30,31 are multiplied by zero A columns: we clamp those
// loads to b=28,29 (valid addresses, values annihilated) -> no divergent
// branches, EXEC all-1s throughout (wave-uniform node loop), as WMMA requires.
// All W offsets fit in 32 bits -> scalar-base + u32 voffset addressing.
__global__ __launch_bounds__(256) void rgcn_basis_wmma(
    const float* __restrict__ W,     // [BB*NN, OUTD]
    const float* __restrict__ Wc,    // [SS, BB]
    float* __restrict__ Vblk,        // [CR*NPR, SS, OUTD]
    int r0, int CR)
{
  __shared__ float lds_wc[SPAD * BPAD];  // zero-padded W_comp, 8 KB
  const int tid = threadIdx.x;
  for (int i = tid; i < SPAD * BPAD; i += 256) {
    int s = i >> 5, b = i & 31;
    lds_wc[i] = (s < SS && b < BB) ? Wc[s * BB + b] : 0.0f;
  }
  __syncthreads();

  const int lane  = tid & 31;
  const int col   = lane & 15;   // output col o / matrix-N index; also A-row M
  const int khalf = lane >> 4;   // 0: K={0,1}, 1: K={2,3} within a 4-wide k-step

  // A operands: 4 m-tiles x 8 k-steps, identical for every node -> registers.
  v2f A[4][8];
  #pragma unroll
  for (int mt = 0; mt < 4; ++mt) {
    #pragma unroll
    for (int kk = 0; kk < 8; ++kk) {
      int srow = 16 * mt + col;
      int kb   = 4 * kk + 2 * khalf;
      A[mt][kk].x = lds_wc[srow * BPAD + kb];
      A[mt][kk].y = lds_wc[srow * BPAD + kb + 1];
    }
  }

  // Lane-constant parts of the 32-bit element offsets into W.
  const unsigned laneA = (unsigned)(2 * khalf) * NN * OUTD + (unsigned)col; // kk 0..6
  const unsigned laneB = (unsigned)col;                                    // kk == 7

  const int wave   = blockIdx.x * 8 + (tid >> 5);
  const int nwaves = gridDim.x * 8;
  const int total  = CR * NPR;

  for (int cn = wave; cn < total; cn += nwaves) {
    const int ng = r0 * NPR + cn;                 // global node id
    const unsigned base = (unsigned)ng * OUTD;
    const unsigned offA = base + laneA;
    const unsigned offB = base + laneB;
    v8f acc[4] = {};
    #pragma unroll
    for (int kk = 0; kk < 8; ++kk) {
      v2f Bm;
      if (kk < 7) {  // b = 4kk + 2khalf, 4kk + 2khalf + 1 (all < 30)
        Bm.x = W[offA + (unsigned)(4 * kk    ) * NN * OUTD];
        Bm.y = W[offA + (unsigned)(4 * kk + 1) * NN * OUTD];
      } else {       // clamp: all lanes load b=28,29; zero A cols kill khalf=1
        Bm.x = W[offB + 28u * NN * OUTD];
        Bm.y = W[offB + 29u * NN * OUTD];
      }
      #pragma unroll
      for (int mt = 0; mt < 4; ++mt) {
        acc[mt] = __builtin_amdgcn_wmma_f32_16x16x4_f32(
            false, A[mt][kk], false, Bm, (short)0, acc[mt], false, false);
      }
    }
    // D layout: VGPR v holds row M = 16*mt + 8*khalf + v, column = col.
    float* vb = Vblk + (size_t)cn * SS * OUTD + col;
    #pragma unroll
    for (int mt = 0; mt < 4; ++mt) {
      #pragma unroll
      for (int v = 0; v < 8; ++v) {
        int s = 16 * mt + 8 * khalf + v;
        if (s < SS) vb[(unsigned)s * OUTD] = acc[mt][v];
      }
    }
  }
}

// Phase 2: per (edge, o): out[src][o] += val * V_slice[dst/50][dst%50][o]
__global__ __launch_bounds__(256) void rgcn_gather_scatter(
    const int* __restrict__ src, const int* __restrict__ dst,
    const float* __restrict__ vals, const float* __restrict__ Vblk,
    float* __restrict__ out, int r0, int CR)
{
  long long gid = (long long)blockIdx.x * 256 + threadIdx.x;
  if (gid >= (long long)CR * EE * OUTD) return;
  unsigned o  = (unsigned)(gid & 15);
  unsigned ei = (unsigned)(gid >> 4);           // < CR*EE <= 1M
  unsigned rl = ei / EE;
  unsigned e  = ei - rl * EE;
  unsigned ge = (unsigned)(r0 + rl) * EE + e;   // < 1M
  int   d = dst[ge];
  int   i = src[ge];
  float v = vals[ge];
  unsigned jj = (unsigned)d / SS, sp = (unsigned)d - jj * SS;
  float x = Vblk[((rl * NPR + jj) * SS + sp) * OUTD + o];  // < 40M, fits u32
  atomicAdd(out + (unsigned)i * OUTD + o, v * x);
}

// Fallback if workspace can't hold even one relation slice: direct per-edge.
__global__ __launch_bounds__(256) void rgcn_direct(
    const int* __restrict__ src, const int* __restrict__ dst,
    const float* __restrict__ vals, const float* __restrict__ W,
    const float* __restrict__ Wc, float* __restrict__ out)
{
  long long gid = (long long)blockIdx.x * 256 + threadIdx.x;
  if (gid >= (long long)SS * EE * OUTD) return;
  unsigned o  = (unsigned)(gid & 15);
  unsigned ei = (unsigned)(gid >> 4);
  unsigned r  = ei / EE;
  unsigned e  = ei - r * EE;
  unsigned ge = r * EE + e;
  int d = dst[ge]; int i = src[ge]; float v = vals[ge];
  unsigned jj = (unsigned)d / SS, sp = (unsigned)d - jj * SS;
  unsigned np = r * NPR + jj;
  float acc = 0.0f;
  #pragma unroll
  for (unsigned b = 0; b < BB; ++b)
    acc = fmaf(Wc[sp * BB + b], W[(b * NN + np) * OUTD + o], acc);
  atomicAdd(out + (unsigned)i * OUTD + o, v * acc);
}

extern "C" void kernel_launch(void* const* d_in, const int* in_sizes, int n_in,
                              void* d_out, int out_size, void* d_ws, size_t ws_size,
                              hipStream_t stream)
{
  // setup_inputs order: features(unused), src, dst, vals, W, W_comp
  const int*   src  = (const int*)  d_in[1];
  const int*   dst  = (const int*)  d_in[2];
  const float* vals = (const float*)d_in[3];
  const float* W    = (const float*)d_in[4];
  const float* Wc   = (const float*)d_in[5];
  float* out = (float*)d_out;

  rgcn_zero<<<(NN * OUTD + 255) / 256, 256, 0, stream>>>(out, NN * OUTD);

  const size_t per_rel = (size_t)NPR * SS * OUTD * sizeof(float);  // 6.4 MB
  int CRmax = (ws_size >= per_rel) ? (int)(ws_size / per_rel) : 0;
  if (CRmax > CRCAP) CRmax = CRCAP;   // keep Vblk chunk L2-resident

  if (CRmax >= 1) {
    for (int r0 = 0; r0 < SS; r0 += CRmax) {
      int CR = (SS - r0 < CRmax) ? (SS - r0) : CRmax;
      int waves  = CR * NPR;              // one wave per node
      int blocks = (waves + 7) / 8;
      rgcn_basis_wmma<<<blocks, 256, 0, stream>>>(W, Wc, (float*)d_ws, r0, CR);
      long long th = (long long)CR * EE * OUTD;
      rgcn_gather_scatter<<<(int)((th + 255) / 256), 256, 0, stream>>>(
          src, dst, vals, (const float*)d_ws, out, r0, CR);
    }
  } else {
    long long th = (long long)SS * EE * OUTD;
    rgcn_direct<<<(int)((th + 255) / 256), 256, 0, stream>>>(src, dst, vals, W, Wc, out);
  }

  rgcn_relu<<<(NN * OUTD + 255) / 256, 256, 0, stream>>>(out, NN * OUTD);
}